// GCNEncoder2_scale_35201551958716
// MI455X (gfx1250) — compile-verified
//
#include <hip/hip_runtime.h>
#include <math.h>

typedef float v2f __attribute__((ext_vector_type(2)));
typedef float v8f __attribute__((ext_vector_type(8)));

// ---------------------------------------------------------------------------
// Degree kernels: deg[i] = 1 (self loop) + in-degree; then dinv = rsqrt(deg)
// ---------------------------------------------------------------------------
__global__ void deg_init_k(float* __restrict__ deg, int N) {
  int i = blockIdx.x * blockDim.x + threadIdx.x;
  if (i < N) deg[i] = 1.0f;
}

__global__ void deg_acc_k(const int* __restrict__ dst, float* __restrict__ deg, int E) {
  int e = blockIdx.x * blockDim.x + threadIdx.x;
  if (e < E) {
    __hip_atomic_fetch_add(&deg[dst[e]], 1.0f, __ATOMIC_RELAXED, __HIP_MEMORY_SCOPE_AGENT);
  }
}

__global__ void dinv_k(float* __restrict__ deg, int N) {
  int i = blockIdx.x * blockDim.x + threadIdx.x;
  if (i < N) {
    float d = deg[i];
    deg[i] = (d > 0.0f) ? rsqrtf(d) : 0.0f;
  }
}

// ---------------------------------------------------------------------------
// Dense GEMM C[M,NN] = A[M,K] @ B[K,NN] using V_WMMA_F32_16X16X4_F32.
// One 16-row M tile per block; wave w owns N-tile w (NN/16 waves per block).
// A frag (16x4): lane<16 -> K=k..k+1, lane>=16 -> K=k+2..k+3, M = lane&15.
// B frag (4x16): same K split, N = n0 + (lane&15).
// C/D: VGPR r, lanes 0-15 -> (M=r, N), lanes 16-31 -> (M=r+8, N).
// ---------------------------------------------------------------------------
template <int NN>
__global__ void gemm_wmma_k(const float* __restrict__ A,
                            const float* __restrict__ B,
                            float* __restrict__ C,
                            int M, int K) {
  const int wave = threadIdx.x >> 5;
  const int lane = threadIdx.x & 31;
  const int m0 = blockIdx.x * 16;
  if (m0 >= M) return;                 // uniform per block: EXEC stays all-1s
  const int n0 = wave * 16;
  const int half = lane >> 4;          // 0 or 1
  const int l15 = lane & 15;

  v8f acc = {};
  const float* aptr = A + (size_t)(m0 + l15) * K + half * 2;
  const float* bptr = B + (size_t)(half * 2) * NN + n0 + l15;

  for (int k = 0; k < K; k += 4) {
    v2f a, b;
    a.x = aptr[0];
    a.y = aptr[1];
    b.x = bptr[0];
    b.y = bptr[NN];
    acc = __builtin_amdgcn_wmma_f32_16x16x4_f32(
        /*neg_a=*/false, a, /*neg_b=*/false, b,
        /*c_mod=*/(short)0, acc, /*reuse_a=*/false, /*reuse_b=*/false);
    aptr += 4;
    bptr += 4 * NN;
  }

  float* cptr = C + (size_t)(m0 + half * 8) * NN + n0 + l15;
#pragma unroll
  for (int r = 0; r < 8; ++r) {
    cptr[0] = acc[r];
    cptr += NN;
  }
}

// ---------------------------------------------------------------------------
// Accumulator init: out[i][c] = dinv[i]^2 * feat[i][c] + bias[c]
// (self-loop contribution + bias, so edge scatter is pure accumulation)
// ---------------------------------------------------------------------------
template <int C>
__global__ void self_bias_k(const float* __restrict__ feat,
                            const float* __restrict__ dinv,
                            const float* __restrict__ bias,
                            float* __restrict__ out, int N) {
  int t = blockIdx.x * blockDim.x + threadIdx.x;
  int total = N * (C / 4);
  if (t >= total) return;
  int i = t / (C / 4);
  int c4 = t % (C / 4);
  float di = dinv[i];
  float w = di * di;
  float4 f = reinterpret_cast<const float4*>(feat)[t];
  float4 bv = reinterpret_cast<const float4*>(bias)[c4];
  float4 o;
  o.x = w * f.x + bv.x;
  o.y = w * f.y + bv.y;
  o.z = w * f.z + bv.z;
  o.w = w * f.w + bv.w;
  reinterpret_cast<float4*>(out)[t] = o;
}

// ---------------------------------------------------------------------------
// Edge scatter: one wave per edge; lanes cover channels.
// out[dst] += dinv[src]*dinv[dst] * feat[src]   (fp32 hardware atomics in L2)
// ---------------------------------------------------------------------------
template <int C>
__global__ void scatter_edges_k(const float* __restrict__ feat,
                                const float* __restrict__ dinv,
                                const int* __restrict__ src,
                                const int* __restrict__ dst,
                                float* __restrict__ out, int E) {
  int wave = blockIdx.x * (blockDim.x >> 5) + (threadIdx.x >> 5);
  int lane = threadIdx.x & 31;
  if (wave >= E) return;
  int s = src[wave];
  int d = dst[wave];
  float norm = dinv[s] * dinv[d];
  const float* frow = feat + (size_t)s * C;
  float* orow = out + (size_t)d * C;
  if constexpr (C == 128) {
    float4 v = reinterpret_cast<const float4*>(frow)[lane];
    float* p = orow + lane * 4;
    __hip_atomic_fetch_add(p + 0, norm * v.x, __ATOMIC_RELAXED, __HIP_MEMORY_SCOPE_AGENT);
    __hip_atomic_fetch_add(p + 1, norm * v.y, __ATOMIC_RELAXED, __HIP_MEMORY_SCOPE_AGENT);
    __hip_atomic_fetch_add(p + 2, norm * v.z, __ATOMIC_RELAXED, __HIP_MEMORY_SCOPE_AGENT);
    __hip_atomic_fetch_add(p + 3, norm * v.w, __ATOMIC_RELAXED, __HIP_MEMORY_SCOPE_AGENT);
  } else {
    float2 v = reinterpret_cast<const float2*>(frow)[lane];
    float* p = orow + lane * 2;
    __hip_atomic_fetch_add(p + 0, norm * v.x, __ATOMIC_RELAXED, __HIP_MEMORY_SCOPE_AGENT);
    __hip_atomic_fetch_add(p + 1, norm * v.y, __ATOMIC_RELAXED, __HIP_MEMORY_SCOPE_AGENT);
  }
}

// ---------------------------------------------------------------------------
// Final per-node min-max scale + L2 normalize (64 feats, one wave32 per node)
// ---------------------------------------------------------------------------
__global__ void finalize_k(const float* __restrict__ g, float* __restrict__ out, int N) {
  int node = blockIdx.x * (blockDim.x >> 5) + (threadIdx.x >> 5);
  int lane = threadIdx.x & 31;
  if (node >= N) return;
  float2 v = reinterpret_cast<const float2*>(g + (size_t)node * 64)[lane];
  float mx = fmaxf(v.x, v.y);
  float mn = fminf(v.x, v.y);
#pragma unroll
  for (int o = 16; o > 0; o >>= 1) {
    mx = fmaxf(mx, __shfl_xor(mx, o, 32));
    mn = fminf(mn, __shfl_xor(mn, o, 32));
  }
  float inv = 1.0f / (mx - mn);        // mirror reference: no eps here
  float sx = (v.x - mn) * inv;
  float sy = (v.y - mn) * inv;
  float ss = sx * sx + sy * sy;
#pragma unroll
  for (int o = 16; o > 0; o >>= 1) ss += __shfl_xor(ss, o, 32);
  float r = 1.0f / fmaxf(sqrtf(ss), 1e-12f);
  float2 ov;
  ov.x = sx * r;
  ov.y = sy * r;
  reinterpret_cast<float2*>(out + (size_t)node * 64)[lane] = ov;
}

// ---------------------------------------------------------------------------
extern "C" void kernel_launch(void* const* d_in, const int* in_sizes, int n_in,
                              void* d_out, int out_size, void* d_ws, size_t ws_size,
                              hipStream_t stream) {
  (void)n_in; (void)out_size; (void)ws_size;

  const float* x  = (const float*)d_in[0];
  const int*   ei = (const int*)d_in[1];
  const float* W1 = (const float*)d_in[2];
  const float* b1 = (const float*)d_in[3];
  const float* W2 = (const float*)d_in[4];
  const float* b2 = (const float*)d_in[5];

  const int N = in_sizes[0] / 128;
  const int E = in_sizes[1] / 2;
  const int* src = ei;
  const int* dst = ei + E;
  float* out = (float*)d_out;

  // workspace: dinv | xw (N*128, reused for layer-2 gemm output) | h (N*128) | g2 (N*64)
  float* ws = (float*)d_ws;
  size_t Npad = ((size_t)N + 63) & ~(size_t)63;
  float* dinv = ws;
  float* xw = ws + Npad;
  float* h  = xw + (size_t)N * 128;
  float* g2 = h  + (size_t)N * 128;

  // degrees -> dinv
  deg_init_k<<<(N + 255) / 256, 256, 0, stream>>>(dinv, N);
  deg_acc_k<<<(E + 255) / 256, 256, 0, stream>>>(dst, dinv, E);
  dinv_k<<<(N + 255) / 256, 256, 0, stream>>>(dinv, N);

  // layer 1: xw = x @ W1 ; h = selfloop+bias ; h += edge scatter
  gemm_wmma_k<128><<<(N + 15) / 16, 256, 0, stream>>>(x, W1, xw, N, 128);
  self_bias_k<128><<<(N * 32 + 255) / 256, 256, 0, stream>>>(xw, dinv, b1, h, N);
  scatter_edges_k<128><<<(E + 7) / 8, 256, 0, stream>>>(xw, dinv, src, dst, h, E);

  // layer 2: z = h @ W2 (into xw region) ; g2 = selfloop+bias ; g2 += scatter
  gemm_wmma_k<64><<<(N + 15) / 16, 128, 0, stream>>>(h, W2, xw, N, 128);
  self_bias_k<64><<<(N * 16 + 255) / 256, 256, 0, stream>>>(xw, dinv, b2, g2, N);
  scatter_edges_k<64><<<(E + 7) / 8, 256, 0, stream>>>(xw, dinv, src, dst, g2, E);

  // min-max scale + L2 normalize
  finalize_k<<<(N + 7) / 8, 256, 0, stream>>>(g2, out, N);
}